// RadiusInteractionGraph_58428735095075
// MI455X (gfx1250) — compile-verified
//
#include <hip/hip_runtime.h>

#define N_NODES 8192
#define MAXNB   32
#define CUTOFF2 100.0f
#define WAVES   4          // waves (=16-row tiles) per block
#define CAP     96         // per-row candidate buffer capacity
#define STRIDE  97         // LDS row stride (bank-conflict avoidance)
#define NK      (N_NODES * MAXNB)

typedef __attribute__((ext_vector_type(2))) float v2f;
typedef __attribute__((ext_vector_type(8))) float v8f;

__global__ void sqnorm_kernel(const float* __restrict__ pos,
                              float* __restrict__ sq, int n) {
    int i = blockIdx.x * blockDim.x + threadIdx.x;
    if (i < n) {
        float x = pos[3 * i + 0], y = pos[3 * i + 1], z = pos[3 * i + 2];
        sq[i] = x * x + y * y + z * z;
    }
}

__global__ __launch_bounds__(WAVES * 32)
void radius_graph_kernel(const float* __restrict__ pos,
                         const int* __restrict__ batch,
                         const float* __restrict__ sq,
                         float* __restrict__ out) {
    __shared__ float s_d2[WAVES][16 * STRIDE];
    __shared__ int   s_ix[WAVES][16 * STRIDE];
    __shared__ int   s_cnt[WAVES][16];

    const int wave = threadIdx.x >> 5;
    const int lane = threadIdx.x & 31;
    const int tile = blockIdx.x * WAVES + wave;
    const int r0   = tile * 16;
    const int n    = lane & 15;   // column-within-tile / M-row for A
    const int hi   = lane >> 4;   // half-wave select (K/M split per ISA layout)

    if (lane < 16) s_cnt[wave][lane] = 0;
    asm volatile("s_wait_dscnt 0x0" ::: "memory");

    // ---- A tile: 16 rows x K=4 (x,y,z,0), ISA 32-bit A layout ----
    v2f a;
    {
        int ra = r0 + n;
        if (hi == 0) { a.x = pos[3 * ra + 0]; a.y = pos[3 * ra + 1]; }
        else         { a.x = pos[3 * ra + 2]; a.y = 0.0f; }
    }

    // per-lane metadata for the 8 C/D rows this lane owns (M = v + 8*hi)
    float sqi[8]; int bi[8];
#pragma unroll
    for (int v = 0; v < 8; ++v) {
        int i  = r0 + v + 8 * hi;
        sqi[v] = sq[i];
        bi[v]  = batch[i];
    }

    // ---- valid column window from sorted batch (uniform binary search) ----
    int blo = batch[r0];
    int bhi = batch[r0 + 15];
    int lo = 0, hiEnd = N_NODES;
    while (lo < hiEnd) { int mid = (lo + hiEnd) >> 1; if (batch[mid] <  blo) lo = mid + 1; else hiEnd = mid; }
    const int colLo = lo;
    lo = 0; hiEnd = N_NODES;
    while (lo < hiEnd) { int mid = (lo + hiEnd) >> 1; if (batch[mid] <= bhi) lo = mid + 1; else hiEnd = mid; }
    const int colHi = lo;

    // ---- scan column tiles: WMMA Gram -> d2 -> filter -> LDS append ----
    for (int j0 = colLo & ~15; j0 < colHi; j0 += 16) {
        if (j0 + 16 < colHi)
            __builtin_prefetch(&pos[3 * (j0 + 16)], 0, 0);   // global_prefetch_b8

        int  j   = j0 + n;
        bool jin = (j < colHi) && (j < N_NODES);
        int  jc  = jin ? j : (N_NODES - 1);

        v2f b;   // B tile: K=4 x 16 cols, mirrored layout
        if (hi == 0) { b.x = pos[3 * jc + 0]; b.y = pos[3 * jc + 1]; }
        else         { b.x = pos[3 * jc + 2]; b.y = 0.0f; }
        float sqj = sq[jc];
        int   bj  = jin ? batch[jc] : -1;

        v8f c = {};
        c = __builtin_amdgcn_wmma_f32_16x16x4_f32(
                /*neg_a=*/false, a, /*neg_b=*/false, b,
                /*c_mod=*/(short)0, c, /*reuse_a=*/false, /*reuse_b=*/false);

#pragma unroll
        for (int v = 0; v < 8; ++v) {
            int   rl = v + 8 * hi;           // local row 0..15
            int   i  = r0 + rl;
            float d2 = sqi[v] + sqj - 2.0f * c[v];
            d2 = fmaxf(d2, 0.0f);
            bool ok = jin && (bi[v] == bj) && (i != j) && (d2 <= CUTOFF2);
            if (ok) {
                int slot = atomicAdd(&s_cnt[wave][rl], 1);   // ds_add_rtn_u32
                if (slot < CAP) {
                    s_d2[wave][rl * STRIDE + slot] = d2;
                    s_ix[wave][rl * STRIDE + slot] = j;
                }
            }
        }
    }
    asm volatile("s_wait_dscnt 0x0" ::: "memory");

    // ---- top-32 selection per row, deterministic (d2, idx) order ----
    if (lane < 16) {
        int r = lane;
        int i = r0 + r;
        int cnt = s_cnt[wave][r]; if (cnt > CAP) cnt = CAP;
        float* d2p = &s_d2[wave][r * STRIDE];
        int*   ixp = &s_ix[wave][r * STRIDE];
        for (int k = 0; k < MAXNB; ++k) {
            float best = 3.0e38f; int bidx = 0x7fffffff; int bslot = -1;
            for (int s = 0; s < cnt; ++s) {
                float d = d2p[s]; int id = ixp[s];
                if (d < best || (d == best && id < bidx)) { best = d; bidx = id; bslot = s; }
            }
            size_t e = (size_t)i * MAXNB + k;
            if (bslot >= 0) {
                d2p[bslot] = 3.4e38f;                 // consume
                out[e]                   = (float)i;     // edge_index[0]
                out[(size_t)NK + e]      = (float)bidx;  // edge_index[1]
                out[2 * (size_t)NK + e]  = sqrtf(best);  // edge_weight
                out[3 * (size_t)NK + e]  = 1.0f;         // mask
            } else {
                out[e]                   = (float)i;
                out[(size_t)NK + e]      = (float)i;     // pad -> self
                out[2 * (size_t)NK + e]  = 0.0f;
                out[3 * (size_t)NK + e]  = 0.0f;
            }
        }
    }
}

extern "C" void kernel_launch(void* const* d_in, const int* in_sizes, int n_in,
                              void* d_out, int out_size, void* d_ws, size_t ws_size,
                              hipStream_t stream) {
    const float* pos   = (const float*)d_in[0];
    const int*   batch = (const int*)d_in[1];
    float*       out   = (float*)d_out;
    float*       sq    = (float*)d_ws;    // 8192 floats = 32 KB scratch

    sqnorm_kernel<<<(N_NODES + 255) / 256, 256, 0, stream>>>(pos, sq, N_NODES);

    const int tiles  = N_NODES / 16;            // 512 row tiles
    const int blocks = tiles / WAVES;           // 128 blocks of 4 waves
    radius_graph_kernel<<<blocks, WAVES * 32, 0, stream>>>(pos, batch, sq, out);
}